// GENConv_62809601737031
// MI455X (gfx1250) — compile-verified
//
#include <hip/hip_runtime.h>
#include <math.h>

#define DIM 64
#define EPSV 1e-7f

typedef __attribute__((ext_vector_type(2))) float v2f;
typedef __attribute__((ext_vector_type(8))) float v8f;

// Order-preserving float <-> uint mapping so atomicMax on u32 == float max
// (works for negatives too, unlike the plain sign trick).
__device__ __forceinline__ unsigned f2ord(float f) {
    unsigned u = __float_as_uint(f);
    return (u & 0x80000000u) ? ~u : (u | 0x80000000u);
}
__device__ __forceinline__ float ord2f(unsigned u) {
    return __uint_as_float((u & 0x80000000u) ? (u & 0x7FFFFFFFu) : ~u);
}

__global__ void genconv_init_ws(unsigned* __restrict__ mx,
                                float* __restrict__ denom,
                                float* __restrict__ numer,
                                int nd) {
    int i = blockIdx.x * blockDim.x + threadIdx.x;
    int stride = gridDim.x * blockDim.x;
    for (; i < nd; i += stride) {
        mx[i]    = 0u;     // minimal ordered key
        denom[i] = 0.0f;
        numer[i] = 0.0f;
    }
}

// Pass 1: one wave32 per edge; lane handles dims (2*lane, 2*lane+1).
// m = relu(node_feats[src] + emb0[ef0] + emb1[ef1]) + eps ; atomicMax(beta*m)
__global__ void genconv_edge_max(const float* __restrict__ nodef,
                                 const float* __restrict__ emb0,
                                 const float* __restrict__ emb1,
                                 const float* __restrict__ beta_p,
                                 const int* __restrict__ src,
                                 const int* __restrict__ dst,
                                 const int* __restrict__ ef0,
                                 const int* __restrict__ ef1,
                                 unsigned* __restrict__ mx,
                                 int n_edges) {
    int e = __builtin_amdgcn_readfirstlane(
        (int)((blockIdx.x * blockDim.x + threadIdx.x) >> 5));
    if (e >= n_edges) return;
    int lane = threadIdx.x & 31;
    float beta = beta_p[0];
    int s = src[e], d = dst[e], f0 = ef0[e], f1 = ef1[e];
    const float2* nf = (const float2*)(nodef + (size_t)s  * DIM);
    const float2* r0 = (const float2*)(emb0  + (size_t)f0 * DIM);
    const float2* r1 = (const float2*)(emb1  + (size_t)f1 * DIM);
    float2 x = nf[lane], a = r0[lane], b = r1[lane];
    float m0 = fmaxf(x.x + a.x + b.x, 0.0f) + EPSV;
    float m1 = fmaxf(x.y + a.y + b.y, 0.0f) + EPSV;
    unsigned* p = mx + (size_t)d * DIM + 2 * lane;
    atomicMax(p,     f2ord(m0 * beta));
    atomicMax(p + 1, f2ord(m1 * beta));
}

// Pass 2: ex = exp(beta*m - mx[dst]); denom += ex; numer += m*ex
__global__ void genconv_edge_sum(const float* __restrict__ nodef,
                                 const float* __restrict__ emb0,
                                 const float* __restrict__ emb1,
                                 const float* __restrict__ beta_p,
                                 const int* __restrict__ src,
                                 const int* __restrict__ dst,
                                 const int* __restrict__ ef0,
                                 const int* __restrict__ ef1,
                                 const unsigned* __restrict__ mx,
                                 float* __restrict__ denom,
                                 float* __restrict__ numer,
                                 int n_edges) {
    int e = __builtin_amdgcn_readfirstlane(
        (int)((blockIdx.x * blockDim.x + threadIdx.x) >> 5));
    if (e >= n_edges) return;
    int lane = threadIdx.x & 31;
    float beta = beta_p[0];
    int s = src[e], d = dst[e], f0 = ef0[e], f1 = ef1[e];
    const float2* nf = (const float2*)(nodef + (size_t)s  * DIM);
    const float2* r0 = (const float2*)(emb0  + (size_t)f0 * DIM);
    const float2* r1 = (const float2*)(emb1  + (size_t)f1 * DIM);
    float2 x = nf[lane], a = r0[lane], b = r1[lane];
    float m0 = fmaxf(x.x + a.x + b.x, 0.0f) + EPSV;
    float m1 = fmaxf(x.y + a.y + b.y, 0.0f) + EPSV;
    const uint2* mr = (const uint2*)(mx + (size_t)d * DIM);
    uint2 mk = mr[lane];
    float ex0 = __expf(m0 * beta - ord2f(mk.x));
    float ex1 = __expf(m1 * beta - ord2f(mk.y));
    size_t off = (size_t)d * DIM + 2 * lane;
    atomicAdd(denom + off,     ex0);
    atomicAdd(denom + off + 1, ex1);
    atomicAdd(numer + off,     m0 * ex0);
    atomicAdd(numer + off + 1, m1 * ex1);
}

// Node pass: msg = numer/denom, MessageNorm, residual. One wave32 per node.
__global__ void genconv_node(const float* __restrict__ nodef,
                             const float* __restrict__ scale_p,
                             const float* __restrict__ denom,
                             const float* __restrict__ numer,
                             float* __restrict__ feats,
                             int n_nodes) {
    int n = __builtin_amdgcn_readfirstlane(
        (int)((blockIdx.x * blockDim.x + threadIdx.x) >> 5));
    if (n >= n_nodes) return;
    int lane = threadIdx.x & 31;
    float scale = scale_p[0];
    const float2* xr = (const float2*)(nodef + (size_t)n * DIM);
    const float2* dr = (const float2*)(denom + (size_t)n * DIM);
    const float2* nr = (const float2*)(numer + (size_t)n * DIM);
    float2 x = xr[lane], de = dr[lane], nu = nr[lane];
    float m0 = (de.x > 0.0f) ? nu.x / de.x : 0.0f;
    float m1 = (de.y > 0.0f) ? nu.y / de.y : 0.0f;
    float pm = m0 * m0 + m1 * m1;
    float pf = x.x * x.x + x.y * x.y;
    #pragma unroll
    for (int off = 16; off > 0; off >>= 1) {   // wave32 reduction
        pm += __shfl_xor(pm, off, 32);
        pf += __shfl_xor(pf, off, 32);
    }
    float r = sqrtf(pf) * scale / fmaxf(sqrtf(pm), 1e-12f);
    float2 o;
    o.x = x.x + m0 * r;
    o.y = x.y + m1 * r;
    ((float2*)(feats + (size_t)n * DIM))[lane] = o;
}

// GEMM: out = feats @ W + b using V_WMMA_F32_16X16X4_F32.
// One wave per 16-row tile; 4 N-tiles of 16 cols; K=64 in steps of 4.
// W is staged in LDS pre-swizzled into (K-pair, N, j) so every B fragment
// is one contiguous ds_load_b64 into an even-aligned VGPR pair.
__global__ __launch_bounds__(256) void genconv_gemm(
        const float* __restrict__ feats,
        const float* __restrict__ W,
        const float* __restrict__ bias,
        float* __restrict__ out,
        int n_nodes) {
    __shared__ float Wl[DIM * DIM];   // Wl[(k>>1)*128 + n*2 + (k&1)] = W[k][n]
    __shared__ float bl[DIM];
    for (int idx = threadIdx.x; idx < DIM * DIM; idx += blockDim.x) {
        int k = idx >> 6;            // row (K index) of W
        int n = idx & 63;            // col (N index)
        Wl[(k >> 1) * 128 + n * 2 + (k & 1)] = W[idx];
    }
    if (threadIdx.x < DIM) bl[threadIdx.x] = bias[threadIdx.x];
    __syncthreads();

    int wave = (int)((blockIdx.x * blockDim.x + threadIdx.x) >> 5);
    int r0 = __builtin_amdgcn_readfirstlane(wave) * 16;
    if (r0 >= n_nodes) return;                  // uniform exit: EXEC stays all-1s

    int lane = threadIdx.x & 31;
    int mrow = lane & 15;
    int hi   = lane >> 4;                       // 0: lanes 0-15, 1: lanes 16-31

    // Clamp tail rows instead of predicating: a duplicated A-row only affects
    // output rows that the guarded store below never writes.
    int row = r0 + mrow;
    if (row >= n_nodes) row = n_nodes - 1;
    const float* arow = feats + (size_t)row * DIM;

    v8f acc0 = {}, acc1 = {}, acc2 = {}, acc3 = {};

    #pragma unroll
    for (int k0 = 0; k0 < DIM; k0 += 4) {
        int ka = k0 + 2 * hi;                   // A: VGPR0=K+0, VGPR1=K+1 (hi lanes +2)
        v2f a = *(const v2f*)(arow + ka);       // unconditional global_load_b64
        // B fragments: pair p = ka/2, element (p, n) = {W[2p][n], W[2p+1][n]}
        const float* bbase = Wl + (size_t)(ka >> 1) * 128 + mrow * 2;
        v2f b0 = *(const v2f*)(bbase +  0);     // N-tile 0: n = mrow
        v2f b1 = *(const v2f*)(bbase + 32);     // N-tile 1: n = 16 + mrow
        v2f b2 = *(const v2f*)(bbase + 64);     // N-tile 2: n = 32 + mrow
        v2f b3 = *(const v2f*)(bbase + 96);     // N-tile 3: n = 48 + mrow
        acc0 = __builtin_amdgcn_wmma_f32_16x16x4_f32(false, a, false, b0, (short)0, acc0, false, false);
        acc1 = __builtin_amdgcn_wmma_f32_16x16x4_f32(false, a, false, b1, (short)0, acc1, false, false);
        acc2 = __builtin_amdgcn_wmma_f32_16x16x4_f32(false, a, false, b2, (short)0, acc2, false, false);
        acc3 = __builtin_amdgcn_wmma_f32_16x16x4_f32(false, a, false, b3, (short)0, acc3, false, false);
    }

    // C/D layout: VGPR i -> M = i (lanes 0-15) / M = 8+i (lanes 16-31), N = lane%16
    int mbase = hi ? 8 : 0;
    #pragma unroll
    for (int i = 0; i < 8; ++i) {
        int rr = r0 + mbase + i;
        if (rr < n_nodes) {
            float* orow = out + (size_t)rr * DIM;
            orow[ 0 + mrow] = acc0[i] + bl[ 0 + mrow];
            orow[16 + mrow] = acc1[i] + bl[16 + mrow];
            orow[32 + mrow] = acc2[i] + bl[32 + mrow];
            orow[48 + mrow] = acc3[i] + bl[48 + mrow];
        }
    }
}

extern "C" void kernel_launch(void* const* d_in, const int* in_sizes, int n_in,
                              void* d_out, int out_size, void* d_ws, size_t ws_size,
                              hipStream_t stream) {
    const float* nodef = (const float*)d_in[0];
    const float* emb0  = (const float*)d_in[1];
    const float* emb1  = (const float*)d_in[2];
    const float* W     = (const float*)d_in[3];
    const float* bias  = (const float*)d_in[4];
    const float* beta  = (const float*)d_in[5];
    const float* scale = (const float*)d_in[6];
    const int* src     = (const int*)d_in[7];
    const int* dst     = (const int*)d_in[8];
    const int* ef0     = (const int*)d_in[9];
    const int* ef1     = (const int*)d_in[10];

    int n_nodes = in_sizes[0] / DIM;
    int n_edges = in_sizes[7];
    size_t nd = (size_t)n_nodes * DIM;

    unsigned* mx = (unsigned*)d_ws;          // region 0: max keys, later reused as feats
    float* denom = (float*)d_ws + nd;        // region 1
    float* numer = (float*)d_ws + 2 * nd;    // region 2
    float* feats = (float*)d_ws;             // aliases mx (safe: mx dead by then)

    int initBlocks = (int)((nd + 255) / 256);
    genconv_init_ws<<<initBlocks, 256, 0, stream>>>(mx, denom, numer, (int)nd);

    int edgeBlocks = (n_edges + 7) / 8;      // 8 waves (8 edges) per 256-thread block
    genconv_edge_max<<<edgeBlocks, 256, 0, stream>>>(
        nodef, emb0, emb1, beta, src, dst, ef0, ef1, mx, n_edges);
    genconv_edge_sum<<<edgeBlocks, 256, 0, stream>>>(
        nodef, emb0, emb1, beta, src, dst, ef0, ef1, mx, denom, numer, n_edges);

    int nodeBlocks = (n_nodes + 7) / 8;      // 8 nodes per block
    genconv_node<<<nodeBlocks, 256, 0, stream>>>(
        nodef, scale, denom, numer, feats, n_nodes);

    int tiles = (n_nodes + 15) / 16;         // one wave per 16-row tile
    int gemmBlocks = (tiles + 7) / 8;
    genconv_gemm<<<gemmBlocks, 256, 0, stream>>>(
        feats, W, bias, (float*)d_out, n_nodes);
}